// BottomUpHTMM_30356828848685
// MI455X (gfx1250) — compile-verified
//
#include <hip/hip_runtime.h>

typedef __attribute__((ext_vector_type(2))) float v2f;
typedef __attribute__((ext_vector_type(8))) float v8f;

#define CDIM 32
#define NDIM 21845
#define LEAF_START 5461
#define NLEAF 16384

__device__ __forceinline__ float waveSum(float v) {
#pragma unroll
  for (int off = 16; off > 0; off >>= 1) v += __shfl_xor(v, off, 32);
  return v;
}

// ---------------------------------------------------------------------------
// Prep: softmaxes + fused tables.
//   Gt[jl*32+i]  = smA[i,j,l]*smSP[l]          (jl = l*32+j)  for upward GEMM
//   G [i*128+jl] = smA[i,j,l]*smSP[l]                          for downward GEMM
//   W [i*128+jl] = G * (log smA[i,j,l] + log smSP[l])          for ll(t_eps) GEMM
//   smB/logB row-major (i,m);  smPi/logPi as [l*32+i]
// ---------------------------------------------------------------------------
__global__ void prep_kernel(const float* __restrict__ A, const float* __restrict__ B,
                            const float* __restrict__ Pi, const float* __restrict__ SP,
                            float* Gt, float* G, float* W, float* smB, float* logB,
                            float* smPi, float* logPi, float* out) {
  __shared__ float sSP[4], sLSP[4];
  int t = threadIdx.x;
  if (t == 0) {
    float m = SP[0];
    for (int l = 1; l < 4; ++l) m = fmaxf(m, SP[l]);
    float s = 0.f;
    for (int l = 0; l < 4; ++l) s += expf(SP[l] - m);
    float ls = logf(s);
    for (int l = 0; l < 4; ++l) { sSP[l] = expf(SP[l] - m) / s; sLSP[l] = SP[l] - m - ls; }
    out[0] = 0.f;
  }
  __syncthreads();
  {  // smA column (j,l): softmax over i (stride 128 in A)
    int j = t >> 2, l = t & 3;
    float m = -1e30f;
    for (int i = 0; i < 32; ++i) m = fmaxf(m, A[(i * 32 + j) * 4 + l]);
    float s = 0.f;
    for (int i = 0; i < 32; ++i) s += expf(A[(i * 32 + j) * 4 + l] - m);
    float ls = logf(s);
    int jl = l * 32 + j;
    for (int i = 0; i < 32; ++i) {
      float a = A[(i * 32 + j) * 4 + l];
      float p = expf(a - m) / s;
      float lp = a - m - ls;
      float g = p * sSP[l];
      G[i * 128 + jl] = g;
      W[i * 128 + jl] = g * (lp + sLSP[l]);
      Gt[jl * 32 + i] = g;
    }
  }
  if (t < 32) {  // smB row i: softmax over m
    int i = t;
    float m = -1e30f;
    for (int mm = 0; mm < 128; ++mm) m = fmaxf(m, B[i * 128 + mm]);
    float s = 0.f;
    for (int mm = 0; mm < 128; ++mm) s += expf(B[i * 128 + mm] - m);
    float ls = logf(s);
    for (int mm = 0; mm < 128; ++mm) {
      float b = B[i * 128 + mm];
      smB[i * 128 + mm] = expf(b - m) / s;
      logB[i * 128 + mm] = b - m - ls;
    }
  }
  if (t < 4) {  // smPi column l: softmax over i
    int l = t;
    float m = -1e30f;
    for (int i = 0; i < 32; ++i) m = fmaxf(m, Pi[i * 4 + l]);
    float s = 0.f;
    for (int i = 0; i < 32; ++i) s += expf(Pi[i * 4 + l] - m);
    float ls = logf(s);
    for (int i = 0; i < 32; ++i) {
      float p = Pi[i * 4 + l];
      smPi[l * 32 + i] = expf(p - m) / s;
      logPi[l * 32 + i] = p - m - ls;
    }
  }
}

// One wave per leaf: beta[n,i] = smPi[i,pos]*smB[i,lab], normalized over i.
__global__ void leaf_kernel(const int* __restrict__ labels, const float* __restrict__ smPi,
                            const float* __restrict__ smB, float* beta) {
  int wave = (blockIdx.x * blockDim.x + threadIdx.x) >> 5;
  int lane = threadIdx.x & 31;
  int n = LEAF_START + wave;
  if (n >= NDIM) return;
  int lpos = (n - LEAF_START) & 3;
  int lab = labels[n];
  float v = smPi[lpos * 32 + lane] * smB[lane * 128 + lab];
  float s = waveSum(v);
  beta[n * 32 + lane] = v / s;
}

// ---------------------------------------------------------------------------
// Upward level: TB(16 x 32) = BetaChFlat(16 x 128) @ Gt(128 x 32) via 16x16x4
// fp32 WMMA; then normalized beta = TB*smB[:,label] / rowsum.
// One wave per 16-parent tile; rows clamped for partial tiles (WMMA EXEC=all 1s).
// ---------------------------------------------------------------------------
__global__ void __launch_bounds__(32)
up_kernel(const int* __restrict__ labels, const float* __restrict__ Gt,
          const float* __restrict__ smB, float* beta, float* t_beta,
          int pStart, int cStart, int n_p) {
  __shared__ float stb[16 * 32];
  int lane = threadIdx.x;
  int tile = blockIdx.x * 16;
  int arow = lane & 15;
  int kpair = (lane >> 4) << 1;  // lanes 0-15 -> K{0,1}, lanes 16-31 -> K{2,3}
  int prow = tile + arow; if (prow > n_p - 1) prow = n_p - 1;
  const float* aBase = beta + (size_t)cStart * 32 + (size_t)prow * 128;  // 4 child rows, contiguous
  int bcol = lane & 15;
  v8f c0 = {}; v8f c1 = {};
#pragma unroll
  for (int kb = 0; kb < 32; ++kb) {
    int k = 4 * kb + kpair;
    v2f a;  a.x = aBase[k];                 a.y = aBase[k + 1];
    v2f b0; b0.x = Gt[k * 32 + bcol];       b0.y = Gt[(k + 1) * 32 + bcol];
    v2f b1; b1.x = Gt[k * 32 + 16 + bcol];  b1.y = Gt[(k + 1) * 32 + 16 + bcol];
    c0 = __builtin_amdgcn_wmma_f32_16x16x4_f32(false, a, false, b0, (short)0, c0, false, false);
    c1 = __builtin_amdgcn_wmma_f32_16x16x4_f32(false, a, false, b1, (short)0, c1, false, false);
  }
  int row_off = (lane >> 4) << 3;  // C/D layout: VGPR r = rows r (lanes<16) / r+8 (lanes>=16)
  int col = lane & 15;
#pragma unroll
  for (int r = 0; r < 8; ++r) {
    int row = r + row_off;
    stb[row * 32 + col] = c0[r];
    stb[row * 32 + 16 + col] = c1[r];
  }
  __syncthreads();
  for (int p = 0; p < 16; ++p) {
    int pp = tile + p;
    if (pp >= n_p) break;
    float tbv = stb[p * 32 + lane];
    t_beta[(size_t)(pStart + pp) * 32 + lane] = tbv;
    int lab = labels[pStart + pp];
    float blv = tbv * smB[lane * 128 + lab];
    float s = waveSum(blv);
    beta[(size_t)(pStart + pp) * 32 + lane] = blv / s;
  }
}

__global__ void root_kernel(const float* __restrict__ beta, float* eps) {
  eps[threadIdx.x] = beta[threadIdx.x];
}

// ---------------------------------------------------------------------------
// Downward level: R(16x32)=eps/t_beta; S = R@G, Wp = R@W  (both 16x32 @ 32x128).
// eps_child[p,jl] = beta_ch[p,jl]*S[p,jl];  ll += sum beta_ch*Wp  (t_eps terms).
// 16 v8f accumulators, 128 wmma per wave.
// ---------------------------------------------------------------------------
__global__ void __launch_bounds__(32)
down_kernel(const float* __restrict__ G, const float* __restrict__ W,
            const float* __restrict__ t_beta, const float* __restrict__ beta,
            float* eps, float* out, int pStart, int cStart, int n_p) {
  int lane = threadIdx.x;
  int tile = blockIdx.x * 16;
  int arow = lane & 15;
  int kpair = (lane >> 4) << 1;
  int prow = tile + arow; if (prow > n_p - 1) prow = n_p - 1;
  const float* eRow = eps + (size_t)(pStart + prow) * 32;
  const float* tRow = t_beta + (size_t)(pStart + prow) * 32;
  int bcol = lane & 15;
  v8f cS[8], cW[8];
#pragma unroll
  for (int nt = 0; nt < 8; ++nt) { cS[nt] = (v8f){}; cW[nt] = (v8f){}; }
#pragma unroll
  for (int kb = 0; kb < 8; ++kb) {
    int k = 4 * kb + kpair;
    v2f a; a.x = eRow[k] / tRow[k]; a.y = eRow[k + 1] / tRow[k + 1];
#pragma unroll
    for (int nt = 0; nt < 8; ++nt) {
      int jl = nt * 16 + bcol;
      v2f bg; bg.x = G[k * 128 + jl]; bg.y = G[(k + 1) * 128 + jl];
      v2f bw; bw.x = W[k * 128 + jl]; bw.y = W[(k + 1) * 128 + jl];
      cS[nt] = __builtin_amdgcn_wmma_f32_16x16x4_f32(false, a, false, bg, (short)0, cS[nt], false, false);
      cW[nt] = __builtin_amdgcn_wmma_f32_16x16x4_f32(false, a, false, bw, (short)0, cW[nt], false, false);
    }
  }
  int row_off = (lane >> 4) << 3;
  int col = lane & 15;
  float ll = 0.f;
#pragma unroll
  for (int r = 0; r < 8; ++r) {
    int p = tile + r + row_off;
    if (p < n_p) {
      const float* bRow = beta + (size_t)cStart * 32 + (size_t)p * 128;
      float* eOut = eps + (size_t)cStart * 32 + (size_t)p * 128;
#pragma unroll
      for (int nt = 0; nt < 8; ++nt) {
        int jl = nt * 16 + col;
        float bch = bRow[jl];
        eOut[jl] = bch * cS[nt][r];
        ll += bch * cW[nt][r];
      }
    }
  }
  ll = waveSum(ll);
  if (lane == 0) atomicAdd(out, ll);
}

// ll += sum_n eps[n,:].logB[:,lab[n]]  (+ leaf: eps.logPi[:,pos]) — one wave/node.
__global__ void ll_eps_kernel(const int* __restrict__ labels, const float* __restrict__ eps,
                              const float* __restrict__ logB, const float* __restrict__ logPi,
                              float* out) {
  int n = (blockIdx.x * blockDim.x + threadIdx.x) >> 5;
  int lane = threadIdx.x & 31;
  if (n >= NDIM) return;
  int lab = labels[n];
  float e = eps[(size_t)n * 32 + lane];
  float acc = e * logB[lane * 128 + lab];
  if (n >= LEAF_START) {
    int lpos = (n - LEAF_START) & 3;
    acc += e * logPi[lpos * 32 + lane];
  }
  float s = waveSum(acc);
  if (lane == 0) atomicAdd(out, s);
}

extern "C" void kernel_launch(void* const* d_in, const int* in_sizes, int n_in,
                              void* d_out, int out_size, void* d_ws, size_t ws_size,
                              hipStream_t stream) {
  const int* labels = (const int*)d_in[0];
  const float* A = (const float*)d_in[1];
  const float* B = (const float*)d_in[2];
  const float* Pi = (const float*)d_in[3];
  const float* SP = (const float*)d_in[4];
  float* out = (float*)d_out;

  float* w = (float*)d_ws;
  float* Gt = w;               // 4096
  float* G = w + 4096;         // 4096
  float* W = w + 8192;         // 4096
  float* smB = w + 12288;      // 4096
  float* logB = w + 16384;     // 4096
  float* smPi = w + 20480;     // 128
  float* logPi = w + 20608;    // 128
  float* beta = w + 20736;     // NDIM*32
  float* t_beta = beta + (size_t)NDIM * 32;
  float* eps = t_beta + (size_t)NDIM * 32;

  static const int starts[9] = {0, 1, 5, 21, 85, 341, 1365, 5461, 21845};

  prep_kernel<<<1, 128, 0, stream>>>(A, B, Pi, SP, Gt, G, W, smB, logB, smPi, logPi, out);
  leaf_kernel<<<NLEAF / 8, 256, 0, stream>>>(labels, smPi, smB, beta);
  for (int d = 6; d >= 0; --d) {
    int n_p = starts[d + 1] - starts[d];
    int blocks = (n_p + 15) / 16;
    up_kernel<<<blocks, 32, 0, stream>>>(labels, Gt, smB, beta, t_beta, starts[d], starts[d + 1], n_p);
  }
  root_kernel<<<1, 32, 0, stream>>>(beta, eps);
  for (int d = 0; d <= 6; ++d) {
    int n_p = starts[d + 1] - starts[d];
    int blocks = (n_p + 15) / 16;
    down_kernel<<<blocks, 32, 0, stream>>>(G, W, t_beta, beta, eps, out, starts[d], starts[d + 1], n_p);
  }
  ll_eps_kernel<<<(NDIM + 7) / 8, 256, 0, stream>>>(labels, eps, logB, logPi, out);
}